// SingleDeformConv_54211077210426
// MI455X (gfx1250) — compile-verified
//
#include <hip/hip_runtime.h>

typedef __attribute__((ext_vector_type(2))) float v2f;
typedef __attribute__((ext_vector_type(8))) float v8f;

#define HH   128
#define WW   128
#define HWSZ (128 * 128)
#define CIN  64
#define COUT 64
#define KCOL (CIN * 9)   // 576 = im2col K dimension, 144 WMMA k-steps of 4
#define SSTR 584         // LDS row stride (not a multiple of 64 banks)
#define TSTR 17

// Fused modulated deformable conv:
//   phase1 im2col -> LDS, phase2 WMMA offsets/mask, phase3 bilinear gather,
//   phase4 WMMA main GEMM, phase5 bias+relu+coalesced store.
__global__ __launch_bounds__(128)
void dcn_fused(const float* __restrict__ data,
               const float* __restrict__ w,
               const float* __restrict__ bias,
               const float* __restrict__ w_off,
               const float* __restrict__ b_off,
               const float* __restrict__ w_mod,
               const float* __restrict__ b_mod,
               float* __restrict__ out)
{
    __shared__ float lds[4 * (16 * SSTR + 32 * TSTR)];

    const int lane = threadIdx.x & 31;
    const int wave = threadIdx.x >> 5;

    float* S  = &lds[wave * (16 * SSTR + 32 * TSTR)];
    float* T  = S + 16 * SSTR;   // 32 x 17 off/mask transpose buffer
    float* TT = S;               // aliased onto S, used only in phase 5

    // wave tile: 16 consecutive pixels along W, 64 output channels
    const int tile  = blockIdx.x * 4 + wave;          // 8192 tiles total
    const int wbase = (tile & 7) * 16;
    const int hrow  = (tile >> 3) & 127;
    const int b     = tile >> 10;

    const int m  = lane & 15;   // pixel row of A / column N of B&D
    const int kh = lane >> 4;   // K-half selector (lanes 16-31 take K= +2,+3)

    const float* dplane = data + (size_t)b * CIN * HWSZ;

    // ---------------- Phase 1: im2col into S ----------------
    #pragma unroll
    for (int kk = 0; kk < 9; ++kk) {
        const int y = hrow + (kk / 3) - 1;
        const int x = wbase + m + (kk % 3) - 1;
        const bool valid = (y >= 0) && (y < HH) && (x >= 0) && (x < WW);
        const int sp = y * WW + x;
        for (int c = kh; c < CIN; c += 2) {
            float v = valid ? dplane[c * HWSZ + sp] : 0.0f;
            S[m * SSTR + c * 9 + kk] = v;
        }
    }

    // ---------------- Phase 2: offsets + modulation GEMM (N=27 pad 32) ----
    v8f accA0 = {};
    v8f accA1 = {};
    {
        const int n1 = 16 + m;                 // channels 16..31 in tile 1
        const float* bp0 = w_off + m * KCOL;   // n0 = m in 0..15 (< 18)
        const float* bp1 = (n1 < 18) ? (w_off + n1 * KCOL)
                                     : (w_mod + (n1 < 27 ? (n1 - 18) : 0) * KCOL);
        const bool z1 = (n1 >= 27);            // zero-padded channels

        for (int t = 0; t < 144; ++t) {
            const int k = t * 4 + 2 * kh;
            v2f a   = *(const v2f*)&S[m * SSTR + k];
            v2f bv0 = *(const v2f*)&bp0[k];
            v2f bv1 = *(const v2f*)&bp1[k];
            if (z1) { bv1.x = 0.0f; bv1.y = 0.0f; }
            accA0 = __builtin_amdgcn_wmma_f32_16x16x4_f32(
                false, a, false, bv0, (short)0, accA0, false, false);
            accA1 = __builtin_amdgcn_wmma_f32_16x16x4_f32(
                false, a, false, bv1, (short)0, accA1, false, false);
        }
    }

    // epilogue A: D tile (M=r+8*kh, N=lane&15) -> T[ch][pixel], bias/sigmoid
    {
        const int ch0 = m;           // 0..15  : offset channels
        const int ch1 = 16 + m;      // 16..31 : 16,17 offset; 18..26 mask; pad
        const float bo0 = b_off[ch0];
        const float bo1 = b_off[ch1 < 18 ? ch1 : 17];
        const int  bmi  = (ch1 >= 18) ? (ch1 - 18 < 9 ? ch1 - 18 : 8) : 0;
        const float bm1 = b_mod[bmi];
        const bool ismask = (ch1 >= 18) && (ch1 < 27);
        #pragma unroll
        for (int r = 0; r < 8; ++r) {
            const int mm = r + 8 * kh;
            T[ch0 * TSTR + mm] = accA0[r] + bo0;
            const float voff = accA1[r] + bo1;
            const float vmsk = 2.0f / (1.0f + __expf(-(accA1[r] + bm1)));
            T[ch1 * TSTR + mm] = ismask ? vmsk : voff;
        }
    }

    // ---------------- Phase 3: bilinear sampling * mask -> S --------------
    #pragma unroll
    for (int kk = 0; kk < 9; ++kk) {
        const float dy = T[(2 * kk) * TSTR + m];
        const float dx = T[(2 * kk + 1) * TSTR + m];
        const float mk = T[(18 + kk) * TSTR + m];
        const float py = (float)(hrow - 1 + kk / 3) + dy;
        const float px = (float)(wbase + m - 1 + kk % 3) + dx;
        const float y0f = floorf(py), x0f = floorf(px);
        const float wy1 = py - y0f, wx1 = px - x0f;
        const float wy0 = 1.0f - wy1, wx0 = 1.0f - wx1;
        const int y0 = (int)y0f, x0 = (int)x0f;
        const int y1 = y0 + 1,  x1 = x0 + 1;
        const bool vy0 = (y0 >= 0) && (y0 < HH), vy1 = (y1 >= 0) && (y1 < HH);
        const bool vx0 = (x0 >= 0) && (x0 < WW), vx1 = (x1 >= 0) && (x1 < WW);
        const int yc0 = min(max(y0, 0), HH - 1), yc1 = min(max(y1, 0), HH - 1);
        const int xc0 = min(max(x0, 0), WW - 1), xc1 = min(max(x1, 0), WW - 1);
        const int i00 = yc0 * WW + xc0, i01 = yc0 * WW + xc1;
        const int i10 = yc1 * WW + xc0, i11 = yc1 * WW + xc1;
        const float g00 = (vy0 && vx0) ? wy0 * wx0 * mk : 0.0f;
        const float g01 = (vy0 && vx1) ? wy0 * wx1 * mk : 0.0f;
        const float g10 = (vy1 && vx0) ? wy1 * wx0 * mk : 0.0f;
        const float g11 = (vy1 && vx1) ? wy1 * wx1 * mk : 0.0f;
        for (int c = kh; c < CIN; c += 2) {
            const float* p = dplane + c * HWSZ;
            float v = g00 * p[i00] + g01 * p[i01] + g10 * p[i10] + g11 * p[i11];
            S[m * SSTR + c * 9 + kk] = v;
        }
    }

    // ---------------- Phase 4: main GEMM  [16 x 576] x [576 x 64] ---------
    v8f acc0 = {}, acc1 = {}, acc2 = {}, acc3 = {};
    {
        const float* bp0 = w + ( 0 + m) * KCOL;
        const float* bp1 = w + (16 + m) * KCOL;
        const float* bp2 = w + (32 + m) * KCOL;
        const float* bp3 = w + (48 + m) * KCOL;
        for (int t = 0; t < 144; ++t) {
            const int k = t * 4 + 2 * kh;
            v2f a = *(const v2f*)&S[m * SSTR + k];   // A frag reused 4x
            acc0 = __builtin_amdgcn_wmma_f32_16x16x4_f32(
                false, a, false, *(const v2f*)&bp0[k], (short)0, acc0, false, false);
            acc1 = __builtin_amdgcn_wmma_f32_16x16x4_f32(
                false, a, false, *(const v2f*)&bp1[k], (short)0, acc1, false, false);
            acc2 = __builtin_amdgcn_wmma_f32_16x16x4_f32(
                false, a, false, *(const v2f*)&bp2[k], (short)0, acc2, false, false);
            acc3 = __builtin_amdgcn_wmma_f32_16x16x4_f32(
                false, a, false, *(const v2f*)&bp3[k], (short)0, acc3, false, false);
        }
    }

    // ---------------- Phase 5: bias + ReLU, coalesced NCHW store ----------
    {
        const float bs0 = bias[ 0 + m];
        const float bs1 = bias[16 + m];
        const float bs2 = bias[32 + m];
        const float bs3 = bias[48 + m];
        #pragma unroll
        for (int r = 0; r < 8; ++r) {
            const int mm = r + 8 * kh;
            float v;
            v = acc0[r] + bs0; TT[( 0 + m) * TSTR + mm] = v > 0.0f ? v : 0.0f;
            v = acc1[r] + bs1; TT[(16 + m) * TSTR + mm] = v > 0.0f ? v : 0.0f;
            v = acc2[r] + bs2; TT[(32 + m) * TSTR + mm] = v > 0.0f ? v : 0.0f;
            v = acc3[r] + bs3; TT[(48 + m) * TSTR + mm] = v > 0.0f ? v : 0.0f;
        }
        float* outp = out + (size_t)b * COUT * HWSZ + hrow * WW + wbase;
        #pragma unroll
        for (int i = 0; i < 32; ++i) {
            const int idx = i * 32 + lane;        // 1024 outputs per wave
            const int n  = idx >> 4;
            const int mm = idx & 15;
            outp[n * HWSZ + mm] = TT[n * TSTR + mm];
        }
    }
}

extern "C" void kernel_launch(void* const* d_in, const int* in_sizes, int n_in,
                              void* d_out, int out_size, void* d_ws, size_t ws_size,
                              hipStream_t stream) {
    const float* data  = (const float*)d_in[0];
    const float* w     = (const float*)d_in[1];
    const float* bias  = (const float*)d_in[2];
    const float* w_off = (const float*)d_in[3];
    const float* b_off = (const float*)d_in[4];
    const float* w_mod = (const float*)d_in[5];
    const float* b_mod = (const float*)d_in[6];
    float* outp = (float*)d_out;

    // 8192 wave-tiles (16 pixels x 64 channels), 4 waves per block
    dim3 grid(2048), block(128);
    hipLaunchKernelGGL(dcn_fused, grid, block, 0, stream,
                       data, w, bias, w_off, b_off, w_mod, b_mod, outp);
}